// GCN_22454089023507
// MI455X (gfx1250) — compile-verified
//
#include <hip/hip_runtime.h>
#include <math.h>

#define NN 50000
#define DHID 128
#define DOUT 40
#define BN_EPS 1e-5f

typedef float v2f __attribute__((ext_vector_type(2)));
typedef float v8f __attribute__((ext_vector_type(8)));

// ================= CSR construction =================
__global__ void k_zero_counts(int* counts) {
    int i = blockIdx.x * blockDim.x + threadIdx.x;
    if (i < NN) counts[i] = 0;
}

__global__ void k_count(const int* __restrict__ dst, int* counts, int E) {
    int e = blockIdx.x * blockDim.x + threadIdx.x;
    if (e < E) atomicAdd(&counts[dst[e]], 1);
}

// single-block exclusive scan: rowptr[0..NN], rowptr[NN] = E
__global__ __launch_bounds__(256) void k_scan(const int* __restrict__ counts, int* rowptr) {
    __shared__ int buf[256];
    __shared__ int carry;
    const int tid = threadIdx.x;
    if (tid == 0) carry = 0;
    __syncthreads();
    for (int base = 0; base < NN; base += 256) {
        int i = base + tid;
        int v = (i < NN) ? counts[i] : 0;
        buf[tid] = v;
        __syncthreads();
#pragma unroll
        for (int off = 1; off < 256; off <<= 1) {
            int t = (tid >= off) ? buf[tid - off] : 0;
            __syncthreads();
            buf[tid] += t;
            __syncthreads();
        }
        int incl = buf[tid];
        int c = carry;
        if (i < NN) rowptr[i] = c + incl - v;
        __syncthreads();
        if (tid == 255) carry += buf[255];
        __syncthreads();
    }
    if (tid == 0) rowptr[NN] = carry;
}

__global__ void k_dinv_copy(const int* __restrict__ counts, const int* __restrict__ rowptr,
                            float* dinv, int* cursor) {
    int i = blockIdx.x * blockDim.x + threadIdx.x;
    if (i < NN) {
        dinv[i] = rsqrtf((float)(counts[i] + 1));  // +1 self-loop; always >= 1
        cursor[i] = rowptr[i];
    }
}

__global__ void k_fill(const int* __restrict__ src, const int* __restrict__ dst,
                       const float* __restrict__ dinv, int* cursor,
                       int* __restrict__ col, float* __restrict__ wgt, int E) {
    int e = blockIdx.x * blockDim.x + threadIdx.x;
    if (e < E) {
        int s = src[e], d = dst[e];
        int pos = atomicAdd(&cursor[d], 1);
        col[pos] = s;
        wgt[pos] = dinv[s] * dinv[d];
    }
}

// ================= fp32 WMMA GEMM: out = A[M,128] @ W[128,NCOLS] + bias =================
// One wave per 16x16 tile; V_WMMA_F32_16X16X4_F32, K=128 in 32 steps.
// Fragment layouts per CDNA5 ISA 7.12.2 (lane m/n = l&15, half = l>>4, K pair 2*half).
// MASKED=false for NCOLS multiple of gridDim.y*16 (no clamp/mask VALU in the hot loop).
template <int NCOLS, int LDW, int LDO, bool MASKED>
__global__ __launch_bounds__(32) void k_gemm_wmma(
    const float* __restrict__ A, const float* __restrict__ W,
    const float* __restrict__ bias, float* __restrict__ out)
{
    const int mtile = blockIdx.x;
    const int ntile = blockIdx.y;
    const int lane  = threadIdx.x & 31;
    const int half  = lane >> 4;
    const int m     = lane & 15;
    const int ncol  = ntile * 16 + (lane & 15);
    // branchless clamp + mask: EXEC stays all-1s around the WMMAs
    const int   ncl = MASKED ? (ncol < NCOLS ? ncol : (NCOLS - 1)) : ncol;
    const float msk = MASKED ? (ncol < NCOLS ? 1.0f : 0.0f) : 1.0f;

    const float* arow = A + (size_t)(mtile * 16 + m) * DHID + half * 2;
    const float* wp   = W + (size_t)(half * 2) * LDW + ncl;

    v8f acc = {0.f, 0.f, 0.f, 0.f, 0.f, 0.f, 0.f, 0.f};
#pragma unroll
    for (int k = 0; k < DHID; k += 4) {           // all offsets compile-time constants
        v2f av = *(const v2f*)(arow + k);
        v2f bv;
        if (MASKED) {
            bv.x = wp[k * LDW]       * msk;
            bv.y = wp[k * LDW + LDW] * msk;
        } else {
            bv.x = wp[k * LDW];
            bv.y = wp[k * LDW + LDW];
        }
        acc = __builtin_amdgcn_wmma_f32_16x16x4_f32(
            false, av, false, bv, (short)0, acc, false, false);
    }

    if (ncol < NCOLS) {
        const float bn = bias[ncol];
        const int rowbase = mtile * 16 + half * 8;
#pragma unroll
        for (int r = 0; r < 8; ++r)
            out[(size_t)(rowbase + r) * LDO + ncol] = acc[r] + bn;
    }
}

// ================= CSR SpMM (gather, no atomics) + fused BN/ReLU =================
// One wave per node: 128-wide accumulator = 4 f32 per lane; neighbor rows are
// 512B coalesced reads (feature matrix is L2-resident on the 192MB L2).
__global__ __launch_bounds__(256) void k_spmm128_bn(
    const int* __restrict__ rowptr, const int* __restrict__ col,
    const float* __restrict__ wgt, const float* __restrict__ dinv,
    const float* __restrict__ t, float* __restrict__ outb,
    const float* __restrict__ g, const float* __restrict__ be,
    const float* __restrict__ m, const float* __restrict__ v)
{
    int wid  = (blockIdx.x * blockDim.x + threadIdx.x) >> 5;
    int lane = threadIdx.x & 31;
    if (wid >= NN) return;

    const float* trow = t + (size_t)wid * DHID;
    float d  = dinv[wid];
    float w0 = d * d;                         // self-loop weight
    float a0 = w0 * trow[lane];
    float a1 = w0 * trow[lane + 32];
    float a2 = w0 * trow[lane + 64];
    float a3 = w0 * trow[lane + 96];

    int beg = rowptr[wid], end = rowptr[wid + 1];
    for (int j = beg; j < end; ++j) {
        int s = col[j];
        float w = wgt[j];
        const float* sr = t + (size_t)s * DHID;
        a0 = fmaf(w, sr[lane],      a0);
        a1 = fmaf(w, sr[lane + 32], a1);
        a2 = fmaf(w, sr[lane + 64], a2);
        a3 = fmaf(w, sr[lane + 96], a3);
    }

    // fused inference BN + ReLU
    float sc0 = g[lane]      * rsqrtf(v[lane]      + BN_EPS);
    float sc1 = g[lane + 32] * rsqrtf(v[lane + 32] + BN_EPS);
    float sc2 = g[lane + 64] * rsqrtf(v[lane + 64] + BN_EPS);
    float sc3 = g[lane + 96] * rsqrtf(v[lane + 96] + BN_EPS);
    float y0 = fmaf(a0 - m[lane],      sc0, be[lane]);
    float y1 = fmaf(a1 - m[lane + 32], sc1, be[lane + 32]);
    float y2 = fmaf(a2 - m[lane + 64], sc2, be[lane + 64]);
    float y3 = fmaf(a3 - m[lane + 96], sc3, be[lane + 96]);

    float* orow = outb + (size_t)wid * DHID;
    orow[lane]      = y0 > 0.f ? y0 : 0.f;
    orow[lane + 32] = y1 > 0.f ? y1 : 0.f;
    orow[lane + 64] = y2 > 0.f ? y2 : 0.f;
    orow[lane + 96] = y3 > 0.f ? y3 : 0.f;
}

// ================= final layer: CSR SpMM (40-wide) + fused log-softmax =================
__global__ __launch_bounds__(256) void k_spmm40_lsm(
    const int* __restrict__ rowptr, const int* __restrict__ col,
    const float* __restrict__ wgt, const float* __restrict__ dinv,
    const float* __restrict__ t, float* __restrict__ outp)
{
    int wid  = (blockIdx.x * blockDim.x + threadIdx.x) >> 5;
    int lane = threadIdx.x & 31;
    if (wid >= NN) return;
    bool hi = lane < (DOUT - 32);             // lanes 0..7 also own col lane+32

    const float* trow = t + (size_t)wid * DOUT;
    float d  = dinv[wid];
    float w0 = d * d;
    float a0 = w0 * trow[lane];
    float a1 = hi ? w0 * trow[lane + 32] : 0.f;

    int beg = rowptr[wid], end = rowptr[wid + 1];
    for (int j = beg; j < end; ++j) {
        int s = col[j];
        float w = wgt[j];
        const float* sr = t + (size_t)s * DOUT;
        a0 = fmaf(w, sr[lane], a0);
        if (hi) a1 = fmaf(w, sr[lane + 32], a1);
    }

    // wave-wide log-softmax over the 40 values held by this wave
    float mx = hi ? fmaxf(a0, a1) : a0;
#pragma unroll
    for (int off = 16; off >= 1; off >>= 1) mx = fmaxf(mx, __shfl_xor(mx, off, 32));
    float s = expf(a0 - mx) + (hi ? expf(a1 - mx) : 0.f);
#pragma unroll
    for (int off = 16; off >= 1; off >>= 1) s += __shfl_xor(s, off, 32);
    float lse = mx + logf(s);

    float* orow = outp + (size_t)wid * DOUT;
    orow[lane] = a0 - lse;
    if (hi) orow[lane + 32] = a1 - lse;
}

extern "C" void kernel_launch(void* const* d_in, const int* in_sizes, int n_in,
                              void* d_out, int out_size, void* d_ws, size_t ws_size,
                              hipStream_t stream) {
    const float* x   = (const float*)d_in[0];
    const int*   ei  = (const int*)d_in[1];
    const int    E   = in_sizes[1] / 2;
    const int*   src = ei;
    const int*   dst = ei + E;
    const float* W0  = (const float*)d_in[2];
    const float* b0  = (const float*)d_in[3];
    const float* W1  = (const float*)d_in[4];
    const float* b1  = (const float*)d_in[5];
    const float* W2  = (const float*)d_in[6];
    const float* b2  = (const float*)d_in[7];
    const float* g0  = (const float*)d_in[8];
    const float* be0 = (const float*)d_in[9];
    const float* m0  = (const float*)d_in[10];
    const float* v0  = (const float*)d_in[11];
    const float* g1  = (const float*)d_in[12];
    const float* be1 = (const float*)d_in[13];
    const float* m1  = (const float*)d_in[14];
    const float* v1  = (const float*)d_in[15];
    float* out = (float*)d_out;

    // ---- workspace carve (256B aligned slabs) ----
    char* p = (char*)d_ws;
    auto carve = [&](size_t bytes) { char* r = p; p += (bytes + 255) & ~(size_t)255; return r; };
    int*   counts = (int*)  carve(sizeof(int)   * NN);
    int*   rowptr = (int*)  carve(sizeof(int)   * (NN + 1));
    int*   cursor = (int*)  carve(sizeof(int)   * NN);
    float* dinv   = (float*)carve(sizeof(float) * NN);
    int*   col    = (int*)  carve(sizeof(int)   * E);
    float* wgt    = (float*)carve(sizeof(float) * E);
    float* A      = (float*)carve(sizeof(float) * (size_t)NN * DHID);
    float* B      = (float*)carve(sizeof(float) * (size_t)NN * DHID);

    const int TB = 256;
    const int nodeB = (NN + TB - 1) / TB;
    const int edgeB = (E + TB - 1) / TB;
    const int waveB = (NN * 32 + TB - 1) / TB;   // one wave per node

    // ---- CSR build (once, reused by all 3 layers) ----
    k_zero_counts<<<nodeB, TB, 0, stream>>>(counts);
    k_count      <<<edgeB, TB, 0, stream>>>(dst, counts, E);
    k_scan       <<<1, 256, 0, stream>>>(counts, rowptr);
    k_dinv_copy  <<<nodeB, TB, 0, stream>>>(counts, rowptr, dinv, cursor);
    k_fill       <<<edgeB, TB, 0, stream>>>(src, dst, dinv, cursor, col, wgt, E);

    dim3 gridH(NN / 16, DHID / 16);             // 3125 x 8, no masking needed
    dim3 gridO(NN / 16, (DOUT + 15) / 16);      // 3125 x 3, masked to 40 cols

    // ---- layer 0 ----
    k_gemm_wmma<DHID, DHID, DHID, false><<<gridH, 32, 0, stream>>>(x, W0, b0, A);
    k_spmm128_bn<<<waveB, TB, 0, stream>>>(rowptr, col, wgt, dinv, A, B, g0, be0, m0, v0);
    // ---- layer 1 ----
    k_gemm_wmma<DHID, DHID, DHID, false><<<gridH, 32, 0, stream>>>(B, W1, b1, A);
    k_spmm128_bn<<<waveB, TB, 0, stream>>>(rowptr, col, wgt, dinv, A, B, g1, be1, m1, v1);
    // ---- layer 2 + log-softmax ----
    k_gemm_wmma<DOUT, DOUT, DOUT, true><<<gridO, 32, 0, stream>>>(B, W2, b2, A);
    k_spmm40_lsm<<<waveB, TB, 0, stream>>>(rowptr, col, wgt, dinv, A, out);
}